// Stocknet_64441689309732
// MI455X (gfx1250) — compile-verified
//
#include <hip/hip_runtime.h>
#include <math.h>

// ---- problem constants (match reference) ----
#define B_   16384
#define S_   5
#define D_   6
#define H_   2
#define HD_  3
#define FF_  2048
#define L_   4
#define NTOK (B_ * S_)   // 81920 token rows

typedef float v2f __attribute__((ext_vector_type(2)));
typedef float v8f __attribute__((ext_vector_type(8)));

// gfx1250 async global->LDS copy path (guarded; falls back to plain copies)
#if defined(__has_builtin)
#if __has_builtin(__builtin_amdgcn_global_load_async_to_lds_b128) && \
    __has_builtin(__builtin_amdgcn_s_wait_asynccnt)
#define HAVE_ASYNC_LDS 1
#endif
#endif
#ifndef HAVE_ASYNC_LDS
#define HAVE_ASYNC_LDS 0
#endif

#if HAVE_ASYNC_LDS
// exact builtin parameter types: v4i in addrspace(1) / addrspace(3)
typedef int v4i_ __attribute__((vector_size(16)));
typedef __attribute__((address_space(1))) v4i_ as1_v4i;
typedef __attribute__((address_space(3))) v4i_ as3_v4i;
#endif

// fp32 WMMA: D(16x16,f32) = A(16x4,f32) * B(4x16,f32) + C
static __device__ __forceinline__ v8f wmma4_f32(v2f a, v2f b, v8f c) {
  return __builtin_amdgcn_wmma_f32_16x16x4_f32(
      /*neg_a=*/false, a, /*neg_b=*/false, b,
      /*c_mod=*/(short)0, c, /*reuse_a=*/false, /*reuse_b=*/false);
}

// =====================================================================
// Kernel 1: per-batch-element MHA + residual + LayerNorm1 (tiny, VALU).
// =====================================================================
__global__ __launch_bounds__(256) void attn_ln1_kernel(
    const float* __restrict__ xin, float* __restrict__ xout,
    const float* __restrict__ Wq, const float* __restrict__ bq,
    const float* __restrict__ Wk, const float* __restrict__ bk,
    const float* __restrict__ Wv, const float* __restrict__ bv,
    const float* __restrict__ Wo, const float* __restrict__ bo,
    const float* __restrict__ g1, const float* __restrict__ be1,
    int layer) {
  int b = blockIdx.x * blockDim.x + threadIdx.x;
  if (b >= B_) return;
  const float* Wql = Wq + layer * D_ * D_;
  const float* Wkl = Wk + layer * D_ * D_;
  const float* Wvl = Wv + layer * D_ * D_;
  const float* Wol = Wo + layer * D_ * D_;
  const float* bql = bq + layer * D_;
  const float* bkl = bk + layer * D_;
  const float* bvl = bv + layer * D_;
  const float* bol = bo + layer * D_;
  const float* g1l = g1 + layer * D_;
  const float* be1l = be1 + layer * D_;

  float x[S_][D_];
  const float* xr = xin + (size_t)b * S_ * D_;
#pragma unroll
  for (int t = 0; t < S_; ++t)
#pragma unroll
    for (int i = 0; i < D_; ++i) x[t][i] = xr[t * D_ + i];

  float q[S_][D_], k[S_][D_], v[S_][D_];
#pragma unroll
  for (int t = 0; t < S_; ++t) {
#pragma unroll
    for (int o = 0; o < D_; ++o) {
      float aq = bql[o], ak = bkl[o], av = bvl[o];
#pragma unroll
      for (int i = 0; i < D_; ++i) {
        float xi = x[t][i];
        aq += xi * Wql[o * D_ + i];
        ak += xi * Wkl[o * D_ + i];
        av += xi * Wvl[o * D_ + i];
      }
      q[t][o] = aq; k[t][o] = ak; v[t][o] = av;
    }
  }

  float ctx[S_][D_];
  const float scale = 0.5773502691896258f;  // 1/sqrt(HD)
#pragma unroll
  for (int h = 0; h < H_; ++h) {
#pragma unroll
    for (int t = 0; t < S_; ++t) {
      float sc[S_];
      float mx = -3.0e38f;
#pragma unroll
      for (int u = 0; u < S_; ++u) {
        float s = 0.f;
#pragma unroll
        for (int d = 0; d < HD_; ++d) s += q[t][h * HD_ + d] * k[u][h * HD_ + d];
        s *= scale;
        sc[u] = s;
        mx = s > mx ? s : mx;
      }
      float se = 0.f;
#pragma unroll
      for (int u = 0; u < S_; ++u) { sc[u] = __expf(sc[u] - mx); se += sc[u]; }
      float inv = 1.0f / se;
#pragma unroll
      for (int d = 0; d < HD_; ++d) {
        float a = 0.f;
#pragma unroll
        for (int u = 0; u < S_; ++u) a += sc[u] * v[u][h * HD_ + d];
        ctx[t][h * HD_ + d] = a * inv;
      }
    }
  }

  // out-proj + residual + LayerNorm1
#pragma unroll
  for (int t = 0; t < S_; ++t) {
    float s[D_];
    float mean = 0.f;
#pragma unroll
    for (int o = 0; o < D_; ++o) {
      float a = bol[o] + x[t][o];
#pragma unroll
      for (int i = 0; i < D_; ++i) a += ctx[t][i] * Wol[o * D_ + i];
      s[o] = a;
      mean += a;
    }
    mean *= (1.0f / D_);
    float var = 0.f;
#pragma unroll
    for (int o = 0; o < D_; ++o) { float d = s[o] - mean; var += d * d; }
    var *= (1.0f / D_);
    float rinv = rsqrtf(var + 1e-5f);
    float* orow = xout + ((size_t)b * S_ + t) * D_;
#pragma unroll
    for (int o = 0; o < D_; ++o) orow[o] = (s[o] - mean) * rinv * g1l[o] + be1l[o];
  }
}

// =====================================================================
// Kernel 2: fused FFN  x_out = LN2(x + relu(x W1^T + b1) W2^T + b2)
// fp32 WMMA pipeline, 8 waves/block, 16 token rows per wave, FF streamed
// in chunks of 16 columns. W2 layer slice staged once per block into LDS
// (async global->LDS when available). No divergence in the hot loop.
// =====================================================================
#define HPAD 18    // h-tile LDS row stride (even -> b64 aligned, bank-spread)
#define W2S  2052  // W2 LDS row stride in dwords (16B-aligned rows, bank-spread)

__global__ __launch_bounds__(256) void ffn_ln2_kernel(
    const float* __restrict__ xin, float* __restrict__ xout,
    const float* __restrict__ W1, const float* __restrict__ b1,
    const float* __restrict__ W2, const float* __restrict__ b2,
    const float* __restrict__ g2, const float* __restrict__ be2,
    int layer) {
  const int lane = threadIdx.x & 31;
  const int wave = threadIdx.x >> 5;
  const int half = lane >> 4;   // 0: lanes 0-15, 1: lanes 16-31
  const int n    = lane & 15;   // N column / A row owned by this lane
  const int tokBase = blockIdx.x * 128 + wave * 16;

  const float* W1l = W1 + layer * FF_ * D_;  // [FF][D]
  const float* b1l = b1 + layer * FF_;
  const float* W2l = W2 + layer * D_ * FF_;  // [D][FF]
  const float* b2l = b2 + layer * D_;
  const float* g2l = g2 + layer * D_;
  const float* be2l = be2 + layer * D_;

  __shared__ __align__(16) float lds_w2[D_ * W2S];       // 49.2 KB
  __shared__ __align__(16) float lds_h[8 * 16 * HPAD];   // 9.2 KB
  __shared__ __align__(16) float lds_y[8 * 16 * 8];      // 4.0 KB
  float* hS = lds_h + wave * 16 * HPAD;
  float* yS = lds_y + wave * 16 * 8;

  // ---- stage W2 layer slice (6 x 2048 fp32) into LDS, once per block ----
  {
    const int tid = threadIdx.x;
#if HAVE_ASYNC_LDS
#pragma unroll
    for (int kk = 0; kk < 12; ++kk) {           // 3072 b128 transfers total
      int i = tid + kk * 256;                   // 0..3071
      int row = i >> 9;                         // /512 b128-per-row
      int c4 = (i & 511) << 2;                  // dword column
      const float* g = W2l + row * FF_ + c4;
      float* l = lds_w2 + row * W2S + c4;
      __builtin_amdgcn_global_load_async_to_lds_b128(
          (as1_v4i*)(g), (as3_v4i*)(l), 0, 0);
    }
    __builtin_amdgcn_s_wait_asynccnt(0);
#else
#pragma unroll
    for (int kk = 0; kk < 12; ++kk) {
      int i = tid + kk * 256;
      int row = i >> 9;
      int c4 = (i & 511) << 2;
      *(float4*)(lds_w2 + row * W2S + c4) = *(const float4*)(W2l + row * FF_ + c4);
    }
#endif
  }
  __syncthreads();

  // ---- load this lane's x row (A operand, K=6 padded to 8) ----
  // A 16x4 layout: lanes0-15 VGPR{0,1}=K{0,1}; lanes16-31 VGPR{0,1}=K{2,3}
  const float* xr = xin + (size_t)(tokBase + n) * D_;
  v2f a0 = *(const v2f*)(xr + 2 * half);  // K-chunk 0..3
  v2f a1 = *(const v2f*)(xr + 4);         // K-chunk 4..7 (K=6,7 are pad)
  a1.x = half ? 0.f : a1.x;
  a1.y = half ? 0.f : a1.y;

  // clamped W2 LDS row for this lane's N column (zeroed via select below)
  const bool nvalid = (n < D_);
  const float* w2sr = lds_w2 + (nvalid ? n : 0) * W2S;

  v8f yacc = {0.f, 0.f, 0.f, 0.f, 0.f, 0.f, 0.f, 0.f};

  for (int ff0 = 0; ff0 < FF_; ff0 += 16) {
    // ---- GEMM1: B[k][nn] = W1[ff0+nn][k] ----
    const float* w1r = W1l + (ff0 + n) * D_;
    v2f bB0 = *(const v2f*)(w1r + 2 * half);  // {W1row[2h], W1row[2h+1]}
    v2f bB1 = *(const v2f*)(w1r + 4);         // {W1row[4], W1row[5]} (h=0 only)
    bB1.x = half ? 0.f : bB1.x;
    bB1.y = half ? 0.f : bB1.y;

    v8f h8 = {0.f, 0.f, 0.f, 0.f, 0.f, 0.f, 0.f, 0.f};
    h8 = wmma4_f32(a0, bB0, h8);
    h8 = wmma4_f32(a1, bB1, h8);

    // ---- bias + relu, stage h tile to LDS (row r+8*half, col n) ----
    float bias = b1l[ff0 + n];
#pragma unroll
    for (int r = 0; r < 8; ++r) {
      float t = h8[r] + bias;
      t = t > 0.f ? t : 0.f;
      hS[(r + 8 * half) * HPAD + n] = t;
    }
    asm volatile("" ::: "memory");  // keep LDS store->load order (DS in-order/wave)

    // ---- GEMM2: y += h(16x16) @ W2chunk^T, K=16 as 4 chunks of 4 ----
#pragma unroll
    for (int c = 0; c < 4; ++c) {
      v2f a2 = *(const v2f*)(hS + n * HPAD + c * 4 + 2 * half);
      v2f bv = *(const v2f*)(w2sr + ff0 + c * 4 + 2 * half);
      bv.x = nvalid ? bv.x : 0.f;  // cndmask, no branch
      bv.y = nvalid ? bv.y : 0.f;
      yacc = wmma4_f32(a2, bv, yacc);
    }
  }

  // ---- epilogue: y tile -> LDS, then per-row residual + b2 + LayerNorm2 ----
  if (nvalid) {
#pragma unroll
    for (int r = 0; r < 8; ++r) yS[(r + 8 * half) * 8 + n] = yacc[r];
  }
  asm volatile("" ::: "memory");

  if (lane < 16) {
    int tok = tokBase + lane;
    const float* xres = xin + (size_t)tok * D_;
    float s[D_];
    float mean = 0.f;
#pragma unroll
    for (int d = 0; d < D_; ++d) {
      s[d] = xres[d] + yS[lane * 8 + d] + b2l[d];
      mean += s[d];
    }
    mean *= (1.0f / D_);
    float var = 0.f;
#pragma unroll
    for (int d = 0; d < D_; ++d) { float t = s[d] - mean; var += t * t; }
    var *= (1.0f / D_);
    float rinv = rsqrtf(var + 1e-5f);
    float* orow = xout + (size_t)tok * D_;
#pragma unroll
    for (int d = 0; d < D_; ++d) orow[d] = (s[d] - mean) * rinv * g2l[d] + be2l[d];
  }
}

// =====================================================================
// Kernel 3: classification head, logits = flat(30) Wfc^T + bfc, softmax2
// =====================================================================
__global__ __launch_bounds__(256) void head_kernel(
    const float* __restrict__ xin, const float* __restrict__ Wfc,
    const float* __restrict__ bfc, float* __restrict__ out) {
  int b = blockIdx.x * blockDim.x + threadIdx.x;
  if (b >= B_) return;
  const float* r = xin + (size_t)b * (S_ * D_);
  float l0 = bfc[0], l1 = bfc[1];
#pragma unroll
  for (int i = 0; i < S_ * D_; ++i) {
    float xi = r[i];
    l0 += xi * Wfc[i];
    l1 += xi * Wfc[S_ * D_ + i];
  }
  float mx = l0 > l1 ? l0 : l1;
  float e0 = __expf(l0 - mx), e1 = __expf(l1 - mx);
  float is = 1.0f / (e0 + e1);
  out[(size_t)b * 2 + 0] = e0 * is;
  out[(size_t)b * 2 + 1] = e1 * is;
}

// =====================================================================
extern "C" void kernel_launch(void* const* d_in, const int* in_sizes, int n_in,
                              void* d_out, int out_size, void* d_ws, size_t ws_size,
                              hipStream_t stream) {
  const float* x   = (const float*)d_in[0];
  const float* Wq  = (const float*)d_in[1];
  const float* bq  = (const float*)d_in[2];
  const float* Wk  = (const float*)d_in[3];
  const float* bk  = (const float*)d_in[4];
  const float* Wv  = (const float*)d_in[5];
  const float* bv  = (const float*)d_in[6];
  const float* Wo  = (const float*)d_in[7];
  const float* bo  = (const float*)d_in[8];
  const float* W1  = (const float*)d_in[9];
  const float* b1  = (const float*)d_in[10];
  const float* W2  = (const float*)d_in[11];
  const float* b2  = (const float*)d_in[12];
  const float* g1  = (const float*)d_in[13];
  const float* be1 = (const float*)d_in[14];
  const float* g2  = (const float*)d_in[15];
  const float* be2 = (const float*)d_in[16];
  const float* Wfc = (const float*)d_in[17];
  const float* bfc = (const float*)d_in[18];

  // ping-pong token buffers in workspace: 2 * 81920*6 floats = ~3.9 MB
  float* ws0 = (float*)d_ws;
  float* ws1 = ws0 + (size_t)NTOK * D_;

  const int attnBlocks = B_ / 256;     // 64
  const int ffnBlocks  = NTOK / 128;   // 640 (8 waves x 16 tokens each)

  for (int l = 0; l < L_; ++l) {
    const float* ain = (l == 0) ? x : ws1;
    attn_ln1_kernel<<<attnBlocks, 256, 0, stream>>>(
        ain, ws0, Wq, bq, Wk, bk, Wv, bv, Wo, bo, g1, be1, l);
    ffn_ln2_kernel<<<ffnBlocks, 256, 0, stream>>>(
        ws0, ws1, W1, b1, W2, b2, g2, be2, l);
  }
  head_kernel<<<attnBlocks, 256, 0, stream>>>(ws1, Wfc, bfc, (float*)d_out);
}